// VFM_27719718928488
// MI455X (gfx1250) — compile-verified
//
#include <hip/hip_runtime.h>
#include <hip/hip_bf16.h>

// Problem constants (match reference)
#define NE_C 150000
#define D_C  256
#define B_C  131072
#define N_C  100000
#define M_C  50000

#define TILE_STRIDE 260   // 260 mod 64 == 4 -> conflict-free ds_load_b64 pattern
#define ND_BLOCKS (B_C / 16)
#define NB_NORM 256

typedef float v2f __attribute__((ext_vector_type(2)));
typedef float v8f __attribute__((ext_vector_type(8)));

// ---------------------------------------------------------------------------
// Kernel A: per-entity pass.
//   biases_all[e] = mu_b + |sc_b| * eps_bias[e]
//   kl_tot[e]     = kl(bias) + sum_d kl(entity)
//   entities_all[e,d] = mu + |sc|*eps   (only if materialize)
// One block (256 threads) per entity; fully coalesced streaming.
// ---------------------------------------------------------------------------
__global__ void __launch_bounds__(256)
vfm_entity_kernel(const float* __restrict__ bias_table,
                  const float* __restrict__ entity_table,
                  const float* __restrict__ eps_bias,
                  const float* __restrict__ eps_entity,
                  float* __restrict__ biases_all,
                  float* __restrict__ kl_tot,
                  float* __restrict__ entities_all,
                  int materialize)
{
    __shared__ float red[256];
    const int e = blockIdx.x;
    const int d = threadIdx.x;

    const float mu  = entity_table[(size_t)e * (2 * D_C) + d];
    const float sc  = fabsf(entity_table[(size_t)e * (2 * D_C) + D_C + d]);
    const float eps = eps_entity[(size_t)e * D_C + d];

    if (materialize)
        entities_all[(size_t)e * D_C + d] = mu + sc * eps;

    // kl_normal_std
    float kl = 0.5f * (sc * sc + mu * mu - 1.0f) - logf(sc);
    red[d] = kl;
    __syncthreads();
    #pragma unroll
    for (int s = 128; s > 0; s >>= 1) {
        if (d < s) red[d] += red[d + s];
        __syncthreads();
    }
    if (d == 0) {
        const float bmu = bias_table[2 * e];
        const float bsc = fabsf(bias_table[2 * e + 1]);
        biases_all[e] = bmu + bsc * eps_bias[e];
        const float klb = 0.5f * (bsc * bsc + bmu * bmu - 1.0f) - logf(bsc);
        kl_tot[e] = red[0] + klb;
    }
}

// ---------------------------------------------------------------------------
// Kernel B: user_norm / item_norm partials.
//   user_norm = sum_b 1/nb_occ[x[b,0]], item_norm = sum_b 1/nb_occ[x[b,1]]
// Fixed grid of NB_NORM blocks -> deterministic tree reduction.
// ---------------------------------------------------------------------------
__global__ void __launch_bounds__(256)
vfm_norm_kernel(const int* __restrict__ x,
                const int* __restrict__ nb_occ,
                float* __restrict__ partials /* 2*NB_NORM */)
{
    __shared__ float su[256];
    __shared__ float si[256];
    const int tid = threadIdx.x;
    float un = 0.0f, itn = 0.0f;
    for (int b = blockIdx.x * blockDim.x + tid; b < B_C; b += gridDim.x * blockDim.x) {
        const int u  = x[2 * b];
        const int it = x[2 * b + 1];
        un  += 1.0f / (float)nb_occ[u];
        itn += 1.0f / (float)nb_occ[it];
    }
    su[tid] = un;
    si[tid] = itn;
    __syncthreads();
    #pragma unroll
    for (int s = 128; s > 0; s >>= 1) {
        if (tid < s) { su[tid] += su[tid + s]; si[tid] += si[tid + s]; }
        __syncthreads();
    }
    if (tid == 0) {
        partials[2 * blockIdx.x]     = su[0];
        partials[2 * blockIdx.x + 1] = si[0];
    }
}

// ---------------------------------------------------------------------------
// Kernel C: fold norm partials into scalars:
//   scalars[0] = N/user_norm, scalars[1] = M/item_norm,
//   scalars[2] = global_bias, scalars[3] = kl_global
// ---------------------------------------------------------------------------
__global__ void vfm_scalar_kernel(const float* __restrict__ partials, int nparts,
                                  const float* __restrict__ gbm,
                                  const float* __restrict__ gbs,
                                  const float* __restrict__ epsg,
                                  float* __restrict__ scalars)
{
    if (threadIdx.x == 0 && blockIdx.x == 0) {
        float un = 0.0f, itn = 0.0f;
        for (int i = 0; i < nparts; ++i) {
            un  += partials[2 * i];
            itn += partials[2 * i + 1];
        }
        const float gm = gbm[0];
        const float gs = fabsf(gbs[0]);
        scalars[0] = (float)N_C / un;
        scalars[1] = (float)M_C / itn;
        scalars[2] = gm + gs * epsg[0];
        scalars[3] = 0.5f * (gs * gs + gm * gm - 1.0f) - logf(gs);
    }
}

// ---------------------------------------------------------------------------
// Kernel D: pair kernel. One wave (32 lanes) per 16 pairs.
//  1) Stage the 16 u-rows and 16 i-rows (256 f32 each) into LDS tiles
//     (float4 coalesced global loads; recompute mu+|sc|*eps if needed).
//  2) pair_emb for the 16 pairs = diag( U(16x256) · I(16x256)^T ) via
//     64 V_WMMA_F32_16X16X4_F32 accumulations, split over TWO accumulators
//     (even/odd K-chunks) to break the serial C-dependency chain.
//     A operand: lane<16 -> row=lane,   K = k0+0..1
//                lane>=16 -> row=lane-16, K = k0+2..3      (ds_load_b64)
//     B operand: same addressing against the I tile.
//  3) Dump accumulators lane-linearly to LDS (2x ds_store_b128, no EXEC
//     games), read the diagonal, finish logits, accumulate KL partial.
// ---------------------------------------------------------------------------
__global__ void __launch_bounds__(32)
vfm_pair_kernel(const int* __restrict__ x,
                const float* __restrict__ entity_table,
                const float* __restrict__ eps_entity,
                const float* __restrict__ entities_all,
                const float* __restrict__ biases_all,
                const float* __restrict__ kl_tot,
                const int* __restrict__ nb_occ,
                const float* __restrict__ scalars,
                float* __restrict__ logits,
                float* __restrict__ kl_partials,
                int materialize)
{
    __shared__ float tile[2][16][TILE_STRIDE];  // [u/i][pair][D], row base 16B aligned
    __shared__ __align__(16) float dmat[32 * 8];  // lane-linear accumulator dump

    const int lane = threadIdx.x;
    const int base = blockIdx.x * 16;

    // All 32 row ids for this tile are the 32 consecutive ints x[2*base ..]:
    // element 2*p+side. One coalesced load, broadcast via readlane.
    const int rowids = x[2 * base + lane];

    // ---- stage 32 rows into LDS ----
    #pragma unroll
    for (int r = 0; r < 32; ++r) {
        const int side = r >> 4;     // 0 = user row, 1 = item row
        const int p    = r & 15;
        const int row_id = __builtin_amdgcn_readlane(rowids, 2 * p + side);
        float* dst = &tile[side][p][0];
        if (materialize) {
            const float4* src = (const float4*)(entities_all + (size_t)row_id * D_C);
            float4 v0 = src[lane];
            float4 v1 = src[lane + 32];
            ((float4*)dst)[lane]      = v0;
            ((float4*)dst)[lane + 32] = v1;
        } else {
            const float4* mup = (const float4*)(entity_table + (size_t)row_id * (2 * D_C));
            const float4* scp = (const float4*)(entity_table + (size_t)row_id * (2 * D_C) + D_C);
            const float4* ep  = (const float4*)(eps_entity  + (size_t)row_id * D_C);
            #pragma unroll
            for (int j = 0; j < 2; ++j) {
                float4 m = mup[lane + 32 * j];
                float4 s = scp[lane + 32 * j];
                float4 e = ep [lane + 32 * j];
                float4 o;
                o.x = m.x + fabsf(s.x) * e.x;
                o.y = m.y + fabsf(s.y) * e.y;
                o.z = m.z + fabsf(s.z) * e.z;
                o.w = m.w + fabsf(s.w) * e.w;
                ((float4*)dst)[lane + 32 * j] = o;
            }
        }
    }
    __syncthreads();

    // ---- WMMA accumulation over K (two independent accumulators) ----
    const int rowA = lane & 15;
    const int hk   = (lane >> 4) * 2;
    const float* uBase = &tile[0][rowA][hk];
    const float* iBase = &tile[1][rowA][hk];

    v8f c0 = {};
    v8f c1 = {};
    #pragma unroll 4
    for (int k0 = 0; k0 < D_C; k0 += 8) {
        v2f a0 = *(const v2f*)(uBase + k0);       // 8B-aligned ds_load_b64
        v2f b0 = *(const v2f*)(iBase + k0);
        v2f a1 = *(const v2f*)(uBase + k0 + 4);
        v2f b1 = *(const v2f*)(iBase + k0 + 4);
        // (neg_a, A, neg_b, B, c_mod, C, reuse_a, reuse_b)
        c0 = __builtin_amdgcn_wmma_f32_16x16x4_f32(false, a0, false, b0,
                                                   (short)0, c0, false, false);
        c1 = __builtin_amdgcn_wmma_f32_16x16x4_f32(false, a1, false, b1,
                                                   (short)0, c1, false, false);
    }
    v8f cs = c0 + c1;

    // ---- dump accumulators lane-linearly, then read diagonal ----
    // dmat[lane*8 + v] = cs[v]; diagonal of D for pair p lives at:
    //   p<8 : lane p,    vgpr p    -> flat 9p
    //   p>=8: lane p+16, vgpr p-8  -> flat 9p + 120
    {
        float4 lo, hi;
        lo.x = cs[0]; lo.y = cs[1]; lo.z = cs[2]; lo.w = cs[3];
        hi.x = cs[4]; hi.y = cs[5]; hi.z = cs[6]; hi.w = cs[7];
        float4* dm = (float4*)dmat;
        dm[lane * 2]     = lo;
        dm[lane * 2 + 1] = hi;
    }
    __syncthreads();

    // ---- finalize logits + KL partial ----
    float klpart = 0.0f;
    if (lane < 16) {
        const float emb = dmat[9 * lane + ((lane >= 8) ? 120 : 0)];
        const int b  = base + lane;
        const int u  = __shfl(rowids, 2 * lane, 32);
        const int it = __shfl(rowids, 2 * lane + 1, 32);
        const float pb = biases_all[u] + biases_all[it];
        logits[b] = scalars[2] + pb + emb;

        const float su = ((u  <= N_C) ? scalars[0] : 0.0f) + ((u  > N_C) ? scalars[1] : 0.0f);
        const float si = ((it <= N_C) ? scalars[0] : 0.0f) + ((it > N_C) ? scalars[1] : 0.0f);
        klpart = kl_tot[u]  * su / (float)nb_occ[u]
               + kl_tot[it] * si / (float)nb_occ[it];
    }
    #pragma unroll
    for (int off = 16; off > 0; off >>= 1)
        klpart += __shfl_down(klpart, off, 32);
    if (lane == 0)
        kl_partials[blockIdx.x] = klpart;
}

// ---------------------------------------------------------------------------
// Kernel E: deterministic final reduction of KL partials.
// ---------------------------------------------------------------------------
__global__ void __launch_bounds__(256)
vfm_finalize_kernel(const float* __restrict__ kl_partials, int n,
                    const float* __restrict__ scalars,
                    float* __restrict__ out_kl)
{
    __shared__ float red[256];
    const int tid = threadIdx.x;
    float s = 0.0f;
    for (int i = tid; i < n; i += 256) s += kl_partials[i];
    red[tid] = s;
    __syncthreads();
    #pragma unroll
    for (int st = 128; st > 0; st >>= 1) {
        if (tid < st) red[tid] += red[tid + st];
        __syncthreads();
    }
    if (tid == 0) out_kl[0] = scalars[3] + red[0];
}

// ---------------------------------------------------------------------------
extern "C" void kernel_launch(void* const* d_in, const int* in_sizes, int n_in,
                              void* d_out, int out_size, void* d_ws, size_t ws_size,
                              hipStream_t stream)
{
    const int*   x        = (const int*)  d_in[0];  // (B,2)
    const float* bias_tab = (const float*)d_in[1];  // (NE,2)
    const float* ent_tab  = (const float*)d_in[2];  // (NE,2D)
    const float* gbm      = (const float*)d_in[3];  // (1,)
    const float* gbs      = (const float*)d_in[4];  // (1,)
    const int*   nb_occ   = (const int*)  d_in[5];  // (NE,)
    const float* eps_bias = (const float*)d_in[6];  // (NE,)
    const float* eps_ent  = (const float*)d_in[7];  // (NE,D)
    const float* eps_g    = (const float*)d_in[8];  // (1,)

    float* ws = (float*)d_ws;
    float* biases_all = ws;                                    // NE
    float* kl_tot     = ws + NE_C;                             // NE
    float* normParts  = ws + 2 * (size_t)NE_C;                 // 2*NB_NORM
    float* scalars    = ws + 2 * (size_t)NE_C + 2 * NB_NORM;   // 4 (pad 8)
    float* klParts    = ws + 2 * (size_t)NE_C + 2 * NB_NORM + 8;  // ND_BLOCKS
    const size_t base_f = 2 * (size_t)NE_C + 2 * NB_NORM + 8 + ND_BLOCKS;
    const size_t full_f = base_f + (size_t)NE_C * D_C;

    const int materialize = (ws_size >= full_f * sizeof(float)) ? 1 : 0;
    float* ents_all = materialize ? (ws + base_f) : ws;  // unused if !materialize

    vfm_entity_kernel<<<NE_C, 256, 0, stream>>>(bias_tab, ent_tab, eps_bias, eps_ent,
                                                biases_all, kl_tot, ents_all, materialize);
    vfm_norm_kernel<<<NB_NORM, 256, 0, stream>>>(x, nb_occ, normParts);
    vfm_scalar_kernel<<<1, 32, 0, stream>>>(normParts, NB_NORM, gbm, gbs, eps_g, scalars);
    vfm_pair_kernel<<<ND_BLOCKS, 32, 0, stream>>>(x, ent_tab, eps_ent, ents_all,
                                                  biases_all, kl_tot, nb_occ, scalars,
                                                  (float*)d_out, klParts, materialize);
    vfm_finalize_kernel<<<1, 256, 0, stream>>>(klParts, ND_BLOCKS, scalars,
                                               (float*)d_out + B_C);
}